// keras_hypernet_v1_37460704756486
// MI455X (gfx1250) — compile-verified
//
#include <hip/hip_runtime.h>
#include <cstdint>

typedef __attribute__((ext_vector_type(2))) float v2f;
typedef __attribute__((ext_vector_type(8))) float v8f;

#define NWT 49600   // columns of Wh2 / generated weight vector length
#define NS  2048    // samples

// ---------------------------------------------------------------------------
// Kernel 1: h_hyper = swish(mu @ Wh1 + bh1)   -> H [2048, 256] in workspace
// ---------------------------------------------------------------------------
__global__ __launch_bounds__(256) void hyper_kernel(
    const float* __restrict__ mu, const float* __restrict__ Wh1,
    const float* __restrict__ bh1, float* __restrict__ H) {
  const int n = blockIdx.x;
  const int c = threadIdx.x;
  float acc = bh1[c];
#pragma unroll
  for (int i = 0; i < 16; ++i)
    acc = fmaf(mu[n * 16 + i], Wh1[i * 256 + c], acc);
  const float s = 1.f / (1.f + __expf(-acc));
  H[n * 256 + c] = acc * s;
}

// ---------------------------------------------------------------------------
// Kernel 2: fused weight-generation (WMMA f32 16x16x4) + per-sample apply.
// Grid: (layer_weight_cols/512, 128 sample tiles). Block: 256 threads (8 waves).
//
// Each WG generates a 16-sample x 512-col chunk of  H@Wh2+bh2  into LDS plus a
// 16-col bias tile, then applies W[m,j,:] . h[m,:] + b[m,j] (+swish).
// K=256 is split over 4 accumulators (independent WMMA chains) and the B
// fragments are software-pipelined one group (16 K) ahead.
// ---------------------------------------------------------------------------
__global__ __launch_bounds__(256) void layer_kernel(
    const float* __restrict__ H, const float* __restrict__ Wh2,
    const float* __restrict__ bh2, const float* __restrict__ hin,
    float* __restrict__ hout,
    int offW, int offB, int din, int dout, int is_last) {
  __shared__ float wstage[33 * 16 * 16];  // 33 tiles x (16 cols x 16 samples) = 33.8KB
  __shared__ float hloc[16 * 128];        // input activations for the sample tile

  const int tid  = threadIdx.x;
  const int lane = tid & 31;
  const int wave = tid >> 5;
  const int nsub = lane & 15;   // column-within-tile / sample-within-tile (A frag)
  const int hi   = lane >> 4;   // upper/lower half of wave (K pairing)

  const int ldin  = (din == 128) ? 7 : 6;
  const int rows  = 512 >> ldin;           // output rows (j) this WG covers
  const int j0    = blockIdx.x * rows;
  const int s0    = blockIdx.y * 16;       // sample tile base
  const int wcol0 = offW + blockIdx.x * 512;

  // Stage input activations for this sample tile.
  for (int i = tid; i < (16 << ldin); i += 256) {
    const int m = i >> ldin;
    const int k = i & (din - 1);
    hloc[m * 128 + k] = hin[(size_t)(s0 + m) * din + k];
  }

  // Preload A fragments (16 samples x K=256) into registers, WMMA A layout:
  // lane<16: M=lane, K=4ks+{0,1};  lane>=16: M=lane-16, K=4ks+{2,3}.
  v2f areg[64];
  {
    const float* ap = H + (size_t)(s0 + nsub) * 256 + 2 * hi;
#pragma unroll
    for (int ks = 0; ks < 64; ++ks) {
      areg[ks].x = ap[4 * ks];
      areg[ks].y = ap[4 * ks + 1];
    }
  }
  __syncthreads();

  // Generate 33 output tiles: t<32 -> weight columns, t==32 -> bias columns.
  // (Reference adds bh2 to the whole generated vector, so both paths add bh2[col].)
  for (int t = wave; t < 33; t += 8) {
    int col;
    if (t < 32) {
      col = wcol0 + t * 16 + nsub;
    } else {
      col = offB + j0 + nsub;
      const int cmax = offB + dout - 1;  // clamp: garbage in unused lanes, stays in-bounds
      col = (col < cmax) ? col : cmax;
    }
    const float* bp = Wh2 + (size_t)(2 * hi) * NWT + col;

    v8f a0 = {}, a1 = {}, a2 = {}, a3 = {};
    // Prologue: fetch B fragments for K-group 0 (K = 16 per group, 4 WMMA steps).
    v2f b0, b1, b2, b3;
    b0.x = bp[0];                 b0.y = bp[(size_t)1 * NWT];
    b1.x = bp[(size_t)4 * NWT];   b1.y = bp[(size_t)5 * NWT];
    b2.x = bp[(size_t)8 * NWT];   b2.y = bp[(size_t)9 * NWT];
    b3.x = bp[(size_t)12 * NWT];  b3.y = bp[(size_t)13 * NWT];

#pragma unroll
    for (int g = 0; g < 16; ++g) {
      v2f n0 = b0, n1 = b1, n2 = b2, n3 = b3;
      const float* np = bp + (size_t)16 * NWT;
      if (g < 15) {  // static after unroll: prefetch next group's B fragments
        n0.x = np[0];                 n0.y = np[(size_t)1 * NWT];
        n1.x = np[(size_t)4 * NWT];   n1.y = np[(size_t)5 * NWT];
        n2.x = np[(size_t)8 * NWT];   n2.y = np[(size_t)9 * NWT];
        n3.x = np[(size_t)12 * NWT];  n3.y = np[(size_t)13 * NWT];
      }
      a0 = __builtin_amdgcn_wmma_f32_16x16x4_f32(false, areg[4 * g + 0], false, b0,
                                                 (short)0, a0, false, false);
      a1 = __builtin_amdgcn_wmma_f32_16x16x4_f32(false, areg[4 * g + 1], false, b1,
                                                 (short)0, a1, false, false);
      a2 = __builtin_amdgcn_wmma_f32_16x16x4_f32(false, areg[4 * g + 2], false, b2,
                                                 (short)0, a2, false, false);
      a3 = __builtin_amdgcn_wmma_f32_16x16x4_f32(false, areg[4 * g + 3], false, b3,
                                                 (short)0, a3, false, false);
      b0 = n0; b1 = n1; b2 = n2; b3 = n3;
      bp = np;
    }
    const v8f acc = (a0 + a1) + (a2 + a3);

    const float bias = bh2[col];
    // D layout: lane holds D[M = v + 8*hi][N = nsub] for v=0..7 -> store n-major,
    // m-minor so 8 contiguous floats per lane => two b128 LDS stores.
    const int base = (t * 16 + nsub) * 16 + hi * 8;
    float4* dst = (float4*)&wstage[base];
    dst[0] = make_float4(acc[0] + bias, acc[1] + bias, acc[2] + bias, acc[3] + bias);
    dst[1] = make_float4(acc[4] + bias, acc[5] + bias, acc[6] + bias, acc[7] + bias);
  }
  __syncthreads();

  // Apply: z[m, j0+jl] = b[m,j] + sum_k W[m,j,k] * h[m,k]; swish unless last layer.
  for (int p = tid; p < 16 * rows; p += 256) {
    const int m = p & 15;
    const int jl = p >> 4;
    float z = wstage[(512 + jl) * 16 + m];  // generated bias (already includes bh2)
    const float* wrow = &wstage[((jl << ldin)) * 16 + m];
    const float* hv = &hloc[m * 128];
#pragma unroll 8
    for (int k = 0; k < din; ++k)
      z = fmaf(wrow[k * 16], hv[k], z);
    const int j = j0 + jl;
    if (is_last) {
      hout[(size_t)(s0 + m) * dout + j] = z;
    } else {
      const float s = 1.f / (1.f + __expf(-z));
      hout[(size_t)(s0 + m) * dout + j] = z * s;
    }
  }
}

// ---------------------------------------------------------------------------
extern "C" void kernel_launch(void* const* d_in, const int* in_sizes, int n_in,
                              void* d_out, int out_size, void* d_ws, size_t ws_size,
                              hipStream_t stream) {
  (void)in_sizes; (void)n_in; (void)out_size; (void)ws_size;
  const float* x   = (const float*)d_in[0];  // [2048, 64]
  const float* mu  = (const float*)d_in[1];  // [2048, 16]
  const float* Wh1 = (const float*)d_in[2];  // [16, 256]
  const float* bh1 = (const float*)d_in[3];  // [256]
  const float* Wh2 = (const float*)d_in[4];  // [256, 49600]
  const float* bh2 = (const float*)d_in[5];  // [49600]
  float* out = (float*)d_out;                // [2048, 64]

  // Workspace layout (4 MB total): H (2MB) | h1 (1MB) | h2 (1MB)
  float* H  = (float*)d_ws;
  float* h1 = H + (size_t)NS * 256;
  float* h2 = h1 + (size_t)NS * 128;

  hyper_kernel<<<NS, 256, 0, stream>>>(mu, Wh1, bh1, H);

  // Layer offsets from SPLIT_SIZES cumsum:
  // W0:[0,8192) b0:[8192,8320) W1:[8320,24704) b1:[24704,24832)
  // W2:[24832,41216) b2:[41216,41344) W3:[41344,49536) b3:[49536,49600)
  layer_kernel<<<dim3(16, 128), 256, 0, stream>>>(H, Wh2, bh2, x,  h1,     0,  8192,  64, 128, 0);
  layer_kernel<<<dim3(32, 128), 256, 0, stream>>>(H, Wh2, bh2, h1, h2,  8320, 24704, 128, 128, 0);
  layer_kernel<<<dim3(32, 128), 256, 0, stream>>>(H, Wh2, bh2, h2, h1, 24832, 41216, 128, 128, 0);
  layer_kernel<<<dim3(16, 128), 256, 0, stream>>>(H, Wh2, bh2, h1, out, 41344, 49536, 128, 64, 1);
}